// RGCN_70566312673746
// MI455X (gfx1250) — compile-verified
//
#include <hip/hip_runtime.h>

typedef __attribute__((ext_vector_type(2))) float v2f;
typedef __attribute__((ext_vector_type(4))) float v4f;
typedef __attribute__((ext_vector_type(8))) float v8f;

#define E_ENT 100000
#define R_REL 64
#define I_CH  256
#define O_CH  256

// ---------------------------------------------------------------------------
// Kernel 1: W_sum[r,o] = sum_i W[r,i,o], stored PAIR-INTERLEAVED:
//   Wpack[(p*O_CH + o)*2 + j] = W_sum[2p + j][o],  p = r>>1, j = r&1
// so the main kernel's B fragment (rows krow, krow+1 at column n) is one
// aligned 8-byte load. (64 blocks x 256 threads; reads coalesced 1KB/iter.)
// ---------------------------------------------------------------------------
__global__ void RGCN_wsum_kernel(const float* __restrict__ W,
                                 float* __restrict__ Wpack) {
    const int r = blockIdx.x;     // 0..63
    const int o = threadIdx.x;    // 0..255
    const float* base = W + (size_t)r * I_CH * O_CH + o;
    float s = 0.0f;
    #pragma unroll 8
    for (int i = 0; i < I_CH; ++i) s += base[(size_t)i * O_CH];
    Wpack[(size_t)(r >> 1) * (2 * O_CH) + o * 2 + (r & 1)] = s;
}

// ---------------------------------------------------------------------------
// Kernel 2: x_sum[e] = sum_j x[e,j]   (one wave32 per row)
// Two coalesced float4 loads per lane (512B per load inst), shfl_xor reduce.
// ---------------------------------------------------------------------------
__global__ void RGCN_xsum_kernel(const float* __restrict__ x,
                                 float* __restrict__ xs) {
    const int wave = threadIdx.x >> 5;
    const int lane = threadIdx.x & 31;
    const int row  = blockIdx.x * 8 + wave;
    if (row >= E_ENT) return;
    const float* p = x + (size_t)row * I_CH;
    v4f a = *(const v4f*)(p + lane * 4);         // cols 0..127
    v4f b = *(const v4f*)(p + 128 + lane * 4);   // cols 128..255
    float s = a.x + a.y + a.z + a.w + b.x + b.y + b.z + b.w;
    #pragma unroll
    for (int off = 16; off > 0; off >>= 1)
        s += __shfl_xor(s, off, 32);
    if (lane == 0) xs[row] = s;
}

// ---------------------------------------------------------------------------
// Kernel 3: out[e,o] = ( (xs[e] * (1/cs[e,:])) @ Wsum ) [e,o]
// One wave per 16-entity row block. A = diag(xs) * recip(cs): 16x64 fp32,
// held as 16 K-chunks of v2f per the ISA 16x4 fp32 A layout
// (v0 = {K=0 | K=2}, v1 = {K=1 | K=3} per half-wave). B fragments are single
// b64 loads from the pair-interleaved Wpack (64KB, L0/L2 resident).
// 16 WMMA f32_16x16x4 accumulate K=64; 16 N-tiles cover O=256.
// ---------------------------------------------------------------------------
__global__ void RGCN_main_kernel(const float* __restrict__ cs,
                                 const float* __restrict__ Wpack,
                                 const float* __restrict__ xs,
                                 float* __restrict__ out) {
    const int wave      = threadIdx.x >> 5;
    const int lane      = threadIdx.x & 31;
    const int row_block = blockIdx.x * (256 >> 5) + wave;
    if (row_block >= E_ENT / 16) return;   // whole-wave exit: EXEC stays all-1s

    const int e_base = row_block * 16;
    const int half   = lane >> 4;   // 0: lanes 0-15, 1: lanes 16-31
    const int m      = lane & 15;   // A row / B-D column within half-wave

    // Fold x_sum into A rows: lane's A row is M = m in both halves.
    const float scale = xs[e_base + m];

    v2f a[16];
    const float* csrow = cs + (size_t)(e_base + m) * R_REL + half * 2;
    #pragma unroll
    for (int k = 0; k < 16; ++k) {
        const float c0 = csrow[4 * k + 0];
        const float c1 = csrow[4 * k + 1];
        a[k].x = scale / c0;
        a[k].y = scale / c1;
    }

    #pragma unroll 1
    for (int nt = 0; nt < 16; ++nt) {
        v8f acc = {};
        #pragma unroll
        for (int k = 0; k < 16; ++k) {
            // B rows (4k + 2*half, +1) at column nt*16+m -> pair p = 2k + half
            const int p = 2 * k + half;
            v2f b = *(const v2f*)(Wpack +
                     ((size_t)p * O_CH + nt * 16 + m) * 2);
            acc = __builtin_amdgcn_wmma_f32_16x16x4_f32(
                /*neg_a=*/false, a[k], /*neg_b=*/false, b,
                /*c_mod=*/(short)0, acc, /*reuse_a=*/false, /*reuse_b=*/false);
        }
        // D layout: VGPR v -> out[e_base + v + half*8][nt*16 + m]
        float* obase = out + (size_t)(e_base + half * 8) * O_CH + nt * 16 + m;
        #pragma unroll
        for (int v = 0; v < 8; ++v)
            obase[(size_t)v * O_CH] = acc[v];
    }
}

// ---------------------------------------------------------------------------
extern "C" void kernel_launch(void* const* d_in, const int* in_sizes, int n_in,
                              void* d_out, int out_size, void* d_ws, size_t ws_size,
                              hipStream_t stream) {
    const float* x  = (const float*)d_in[0];   // (E, I)  f32
    const float* cs = (const float*)d_in[1];   // (E, R)  f32
    const float* W  = (const float*)d_in[2];   // (R, I, O) f32
    // d_in[3] = edge_index -- unused by the reference output.
    float* out = (float*)d_out;                // (E, O) f32

    float* Wpack = (float*)d_ws;               // 64*256 floats = 64 KB
    float* xsum  = Wpack + R_REL * O_CH;       // 100000 floats

    RGCN_wsum_kernel<<<R_REL, O_CH, 0, stream>>>(W, Wpack);
    RGCN_xsum_kernel<<<(E_ENT + 7) / 8, 256, 0, stream>>>(x, xsum);

    const int row_blocks = E_ENT / 16;                 // 6250
    const int grid = (row_blocks + 7) / 8;             // 8 waves per block
    RGCN_main_kernel<<<grid, 256, 0, stream>>>(cs, Wpack, xsum, out);
}